// Net_orig_28930899705994
// MI455X (gfx1250) — compile-verified
//
#include <hip/hip_runtime.h>
#include <hip/hip_bf16.h>
#include <math.h>

// ---------------------------------------------------------------------------
// GCN 2-layer inference for gfx1250 (MI455X, wave32, WMMA).
// FP32 matmuls via V_WMMA_F32_16X16X4_F32 (keeps reference fp32 accuracy).
// Graph aggregation via global_atomic_add_f32 scatter (L2-resident features).
// Workspace need: ~119 MB (dinv + h1 + out1 + h2).
// ---------------------------------------------------------------------------

typedef __attribute__((ext_vector_type(2))) float v2f;
typedef __attribute__((ext_vector_type(8))) float v8f;

#define N_FEAT   512
#define HIDDEN   128
#define N_CLS    40

// ------------------------------ degree / norm ------------------------------

__global__ void deg_init_kernel(float* __restrict__ deg, int n) {
    int i = blockIdx.x * blockDim.x + threadIdx.x;
    if (i < n) deg[i] = 1.0f;              // self-loop contributes 1
}

__global__ void deg_edges_kernel(const long long* __restrict__ dst,
                                 float* __restrict__ deg, int e) {
    int i = blockIdx.x * blockDim.x + threadIdx.x;
    if (i < e) atomicAdd(&deg[(int)dst[i]], 1.0f);
}

__global__ void deg_finalize_kernel(float* __restrict__ deg, int n) {
    int i = blockIdx.x * blockDim.x + threadIdx.x;
    if (i < n) deg[i] = rsqrtf(deg[i]);    // deg >= 1 always (self-loop)
}

// ------------------------------ GEMM1: X @ W1 ------------------------------
// M=n_nodes, K=512, N=128. One wave per 16-row M-tile, 8 N-tile accumulators.
// A layout (32-bit 16x4): lanes 0-15 row M=lane, VGPR0=K0,VGPR1=K1;
//                         lanes 16-31 same rows, VGPR0=K2,VGPR1=K3.
// B layout (32-bit 4x16): symmetric — lane half selects K block, col = lane&15.

__global__ void __launch_bounds__(64)
gemm1_wmma_kernel(const float* __restrict__ X, const float* __restrict__ W,
                  float* __restrict__ H, int M) {
    const int wave = threadIdx.x >> 5;
    const int lane = threadIdx.x & 31;
    const int mt   = blockIdx.x * 2 + wave;          // M-tile index
    if (mt * 16 + 16 > M) return;                    // whole-wave uniform guard

    const int row  = mt * 16 + (lane & 15);
    const int koff = (lane >> 4) * 2;
    const int ncol = lane & 15;

    v8f acc[8];
    #pragma unroll
    for (int t = 0; t < 8; ++t)
        #pragma unroll
        for (int i = 0; i < 8; ++i) acc[t][i] = 0.0f;

    const float* __restrict__ xrow = X + (size_t)row * N_FEAT;

    for (int k = 0; k < N_FEAT; k += 4) {
        v2f a;
        a.x = xrow[k + koff];
        a.y = xrow[k + koff + 1];
        #pragma unroll
        for (int t = 0; t < 8; ++t) {
            const float* __restrict__ wp = W + (size_t)(k + koff) * HIDDEN + t * 16 + ncol;
            v2f b;
            b.x = wp[0];
            b.y = wp[HIDDEN];
            acc[t] = __builtin_amdgcn_wmma_f32_16x16x4_f32(
                false, a, false, b, (short)0, acc[t], false, false);
        }
    }

    // C/D layout: VGPR r -> M = r + 8*(lane>=16), N = lane&15
    const int r0 = mt * 16 + (lane >> 4) * 8;
    #pragma unroll
    for (int t = 0; t < 8; ++t)
        #pragma unroll
        for (int r = 0; r < 8; ++r)
            H[(size_t)(r0 + r) * HIDDEN + t * 16 + ncol] = acc[t][r];
}

// --------------------- layer-1 bias + self-loop + scatter ------------------

__global__ void init1_kernel(const float* __restrict__ h1,
                             const float* __restrict__ dinv,
                             const float* __restrict__ b1,
                             float* __restrict__ out1, int n) {
    int idx = blockIdx.x * blockDim.x + threadIdx.x;
    if (idx >= n * HIDDEN) return;
    int node = idx >> 7;          // /128
    int f    = idx & (HIDDEN - 1);
    float d  = dinv[node];
    out1[idx] = b1[f] + h1[idx] * d * d;   // bias + self-loop (norm = dinv^2)
}

__global__ void scatter1_kernel(const long long* __restrict__ src,
                                const long long* __restrict__ dst,
                                const float* __restrict__ dinv,
                                const float* __restrict__ h1,
                                float* __restrict__ out1, int e) {
    int wid  = (blockIdx.x * blockDim.x + threadIdx.x) >> 5;
    int lane = threadIdx.x & 31;
    if (wid >= e) return;
    int s = (int)src[wid];
    int d = (int)dst[wid];
    float w = dinv[s] * dinv[d];
    const float4* __restrict__ hp = (const float4*)(h1 + (size_t)s * HIDDEN);
    float4 v = hp[lane];
    float* __restrict__ op = out1 + (size_t)d * HIDDEN + lane * 4;
    atomicAdd(op + 0, v.x * w);
    atomicAdd(op + 1, v.y * w);
    atomicAdd(op + 2, v.z * w);
    atomicAdd(op + 3, v.w * w);
}

// --------------------- GEMM2: relu(out1) @ W2  (N=40 -> 3 tiles) -----------

__global__ void __launch_bounds__(64)
gemm2_wmma_kernel(const float* __restrict__ Hin, const float* __restrict__ W,
                  float* __restrict__ H2, int M) {
    const int wave = threadIdx.x >> 5;
    const int lane = threadIdx.x & 31;
    const int mt   = blockIdx.x * 2 + wave;
    if (mt * 16 + 16 > M) return;

    const int row  = mt * 16 + (lane & 15);
    const int koff = (lane >> 4) * 2;
    const int ncol = lane & 15;

    v8f acc[3];
    #pragma unroll
    for (int t = 0; t < 3; ++t)
        #pragma unroll
        for (int i = 0; i < 8; ++i) acc[t][i] = 0.0f;

    const float* __restrict__ arow = Hin + (size_t)row * HIDDEN;

    for (int k = 0; k < HIDDEN; k += 4) {
        v2f a;
        a.x = fmaxf(arow[k + koff], 0.0f);       // fused ReLU
        a.y = fmaxf(arow[k + koff + 1], 0.0f);
        #pragma unroll
        for (int t = 0; t < 3; ++t) {
            int col = t * 16 + ncol;
            v2f b;
            b.x = (col < N_CLS) ? W[(size_t)(k + koff) * N_CLS + col] : 0.0f;
            b.y = (col < N_CLS) ? W[(size_t)(k + koff + 1) * N_CLS + col] : 0.0f;
            acc[t] = __builtin_amdgcn_wmma_f32_16x16x4_f32(
                false, a, false, b, (short)0, acc[t], false, false);
        }
    }

    const int r0 = mt * 16 + (lane >> 4) * 8;
    #pragma unroll
    for (int t = 0; t < 3; ++t) {
        int col = t * 16 + ncol;
        if (col < N_CLS) {
            #pragma unroll
            for (int r = 0; r < 8; ++r)
                H2[(size_t)(r0 + r) * N_CLS + col] = acc[t][r];
        }
    }
}

// --------------------- layer-2 bias + self-loop + scatter ------------------

__global__ void init2_kernel(const float* __restrict__ h2,
                             const float* __restrict__ dinv,
                             const float* __restrict__ b2,
                             float* __restrict__ out2, int n) {
    int idx = blockIdx.x * blockDim.x + threadIdx.x;
    if (idx >= n * N_CLS) return;
    int node = idx / N_CLS;
    int c    = idx - node * N_CLS;
    float d  = dinv[node];
    out2[idx] = b2[c] + h2[idx] * d * d;
}

__global__ void scatter2_kernel(const long long* __restrict__ src,
                                const long long* __restrict__ dst,
                                const float* __restrict__ dinv,
                                const float* __restrict__ h2,
                                float* __restrict__ out2, int e) {
    int wid  = (blockIdx.x * blockDim.x + threadIdx.x) >> 5;
    int lane = threadIdx.x & 31;
    if (wid >= e) return;
    int s = (int)src[wid];
    int d = (int)dst[wid];
    float w = dinv[s] * dinv[d];
    const float* __restrict__ hp = h2 + (size_t)s * N_CLS;
    float* __restrict__ op = out2 + (size_t)d * N_CLS;
    atomicAdd(op + lane, hp[lane] * w);
    if (lane < N_CLS - 32)
        atomicAdd(op + lane + 32, hp[lane + 32] * w);
}

// ------------------------------ log-softmax --------------------------------
// One wave per node; lane covers class c=lane and c=lane+32 (if < 40).

__global__ void log_softmax_kernel(float* __restrict__ out, int n) {
    int wid  = (blockIdx.x * blockDim.x + threadIdx.x) >> 5;
    int lane = threadIdx.x & 31;
    if (wid >= n) return;
    float* __restrict__ p = out + (size_t)wid * N_CLS;
    float v0 = p[lane];
    float v1 = (lane < N_CLS - 32) ? p[lane + 32] : -INFINITY;

    float m = fmaxf(v0, v1);
    #pragma unroll
    for (int off = 16; off > 0; off >>= 1)
        m = fmaxf(m, __shfl_xor(m, off, 32));

    float s = __expf(v0 - m) + ((lane < N_CLS - 32) ? __expf(v1 - m) : 0.0f);
    #pragma unroll
    for (int off = 16; off > 0; off >>= 1)
        s += __shfl_xor(s, off, 32);

    float lse = m + logf(s);
    p[lane] = v0 - lse;
    if (lane < N_CLS - 32) p[lane + 32] = v1 - lse;
}

// ------------------------------ host launcher ------------------------------

extern "C" void kernel_launch(void* const* d_in, const int* in_sizes, int n_in,
                              void* d_out, int out_size, void* d_ws, size_t ws_size,
                              hipStream_t stream) {
    (void)n_in; (void)out_size; (void)ws_size;

    const float*     x   = (const float*)d_in[0];
    const long long* ei  = (const long long*)d_in[1];   // int64 [2, E]
    const float*     W1  = (const float*)d_in[2];
    const float*     b1  = (const float*)d_in[3];
    const float*     W2  = (const float*)d_in[4];
    const float*     b2  = (const float*)d_in[5];
    float* out = (float*)d_out;

    const int N = in_sizes[0] / N_FEAT;   // 100000 nodes
    const int E = in_sizes[1] / 2;        // 1600000 edges
    const long long* srcI = ei;
    const long long* dstI = ei + E;

    // Workspace carve-up (256B aligned slices).
    char* ws = (char*)d_ws;
    size_t off = 0;
    auto alloc = [&](size_t bytes) {
        char* p = ws + off;
        off += (bytes + 255) & ~(size_t)255;
        return p;
    };
    float* dinv = (float*)alloc((size_t)N * sizeof(float));
    float* h1   = (float*)alloc((size_t)N * HIDDEN * sizeof(float));
    float* out1 = (float*)alloc((size_t)N * HIDDEN * sizeof(float));
    float* h2   = (float*)alloc((size_t)N * N_CLS * sizeof(float));

    const int TB = 256;
    const int mtiles = N / 16;                     // 6250 (exact)
    const int gemm_blocks = (mtiles + 1) / 2;      // 2 waves of 32 per block

    // 1. degrees -> dinv
    deg_init_kernel<<<(N + TB - 1) / TB, TB, 0, stream>>>(dinv, N);
    deg_edges_kernel<<<(E + TB - 1) / TB, TB, 0, stream>>>(dstI, dinv, E);
    deg_finalize_kernel<<<(N + TB - 1) / TB, TB, 0, stream>>>(dinv, N);

    // 2. h1 = X @ W1  (WMMA fp32)
    gemm1_wmma_kernel<<<gemm_blocks, 64, 0, stream>>>(x, W1, h1, N);

    // 3. out1 = b1 + dinv^2 * h1 (self-loop) + scatter over edges
    init1_kernel<<<((size_t)N * HIDDEN + TB - 1) / TB, TB, 0, stream>>>(h1, dinv, b1, out1, N);
    scatter1_kernel<<<((size_t)E * 32 + TB - 1) / TB, TB, 0, stream>>>(srcI, dstI, dinv, h1, out1, E);

    // 4. h2 = relu(out1) @ W2  (WMMA fp32, ReLU fused in A load)
    gemm2_wmma_kernel<<<gemm_blocks, 64, 0, stream>>>(out1, W2, h2, N);

    // 5. out = b2 + dinv^2 * h2 + scatter over edges
    init2_kernel<<<((size_t)N * N_CLS + TB - 1) / TB, TB, 0, stream>>>(h2, dinv, b2, out, N);
    scatter2_kernel<<<((size_t)E * 32 + TB - 1) / TB, TB, 0, stream>>>(srcI, dstI, dinv, h2, out, E);

    // 6. row-wise log-softmax in place
    log_softmax_kernel<<<((size_t)N * 32 + TB - 1) / TB, TB, 0, stream>>>(out, N);
}